// PCANormSVDPI_37151467110669
// MI455X (gfx1250) — compile-verified
//
#include <hip/hip_runtime.h>

// PCA-whitening (pcanorm) for B=1024 batches of (C=32, M=1024) fp32.
// One workgroup (256 threads = 8 wave32) per batch.
// GEMMs use V_WMMA_F32_16X16X4_F32 (fp32 matches reference precision).

typedef __attribute__((ext_vector_type(2))) float v2f;
typedef __attribute__((ext_vector_type(8))) float v8f;

#define C_DIM 32
#define M_DIM 1024
#define EPSV 1e-5f
#define NUM_ITER 19
#define CP 33          // padded LDS row stride for 32x32 matrices

__device__ __forceinline__ float wave_reduce_add(float v) {
#pragma unroll
  for (int off = 16; off > 0; off >>= 1) v += __shfl_xor(v, off, 32);
  return v;
}

__global__ __launch_bounds__(256, 2)
void pcanorm_kernel(const float* __restrict__ x,
                    const float* __restrict__ v0,
                    float* __restrict__ z) {
  __shared__ float sCov[C_DIM * CP];   // covariance (deflated in place)
  __shared__ float sS[C_DIM * CP];     // S_hat rows
  __shared__ float sMu[C_DIM];         // per-row mean
  __shared__ float sV[C_DIM];          // power-iteration vector broadcast
  __shared__ float sRed[256];

  const int tid  = threadIdx.x;
  const int lane = tid & 31;
  const int wave = tid >> 5;
  const int b    = blockIdx.x;
  const float* __restrict__ xb = x + (size_t)b * (C_DIM * M_DIM);

  // zero the cov accumulator (atomically accumulated below)
  for (int i = tid; i < C_DIM * CP; i += 256) sCov[i] = 0.0f;

  // ---- Step A: per-row mean (8 threads per row) ----
  {
    const int row = tid >> 3;
    const int sub = tid & 7;
    const float* xr = xb + row * M_DIM;
    float s = 0.0f;
#pragma unroll 8
    for (int i = 0; i < M_DIM / 8; ++i) s += xr[i * 8 + sub];
    sRed[tid] = s;
  }
  __syncthreads();
  if (tid < C_DIM) {
    float s = 0.0f;
#pragma unroll
    for (int i = 0; i < 8; ++i) s += sRed[tid * 8 + i];
    sMu[tid] = s * (1.0f / M_DIM);
  }
  __syncthreads();

  // ---- Step B: cov = xc*xc^T / M via V_WMMA_F32_16X16X4_F32 ----
  // fp32 A 16x4 layout: lanes 0-15 hold (M=lane, K+0/K+1), lanes 16-31 (M=lane-16, K+2/K+3).
  // B 4x16 is the mirrored layout, so for a symmetric product A-frag(t) == B-frag(t).
  {
    const int rlo  = lane & 15;
    const int koff = (lane >> 4) * 2;
    const float mu0 = sMu[rlo];
    const float mu1 = sMu[rlo + 16];
    v8f acc00 = {}, acc01 = {}, acc10 = {}, acc11 = {};
    const int kbase = wave * (M_DIM / 8);          // each wave owns a 128-wide K slice
    for (int ks = 0; ks < (M_DIM / 8) / 4; ++ks) {
      const int k = kbase + ks * 4 + koff;
      v2f f0, f1;
      f0.x = xb[rlo * M_DIM + k]            - mu0;
      f0.y = xb[rlo * M_DIM + k + 1]        - mu0;
      f1.x = xb[(rlo + 16) * M_DIM + k]     - mu1;
      f1.y = xb[(rlo + 16) * M_DIM + k + 1] - mu1;
      acc00 = __builtin_amdgcn_wmma_f32_16x16x4_f32(false, f0, false, f0, (short)0, acc00, false, false);
      acc01 = __builtin_amdgcn_wmma_f32_16x16x4_f32(false, f0, false, f1, (short)0, acc01, false, false);
      acc10 = __builtin_amdgcn_wmma_f32_16x16x4_f32(false, f1, false, f0, (short)0, acc10, false, false);
      acc11 = __builtin_amdgcn_wmma_f32_16x16x4_f32(false, f1, false, f1, (short)0, acc11, false, false);
    }
    // C/D layout: VGPR i -> M=i (lanes 0-15) / M=i+8 (lanes 16-31), N = lane&15
    const int mrow = (lane >> 4) ? 8 : 0;
    const int ncol = lane & 15;
    const float sc = 1.0f / M_DIM;
#pragma unroll
    for (int i = 0; i < 8; ++i) {
      atomicAdd(&sCov[(mrow + i) * CP + ncol],           acc00[i] * sc);
      atomicAdd(&sCov[(mrow + i) * CP + ncol + 16],      acc01[i] * sc);
      atomicAdd(&sCov[(mrow + i + 16) * CP + ncol],      acc10[i] * sc);
      atomicAdd(&sCov[(mrow + i + 16) * CP + ncol + 16], acc11[i] * sc);
    }
  }
  __syncthreads();

  // ---- Steps C & D: OAS shrinkage + 32 deflation steps (wave 0 only) ----
  // Each lane owns one row of cov; cross-lane reductions via wave32 shfl_xor.
  if (tid < 32) {
    const int r = lane;
    float trace = wave_reduce_add(sCov[r * CP + r]) * (1.0f / C_DIM);
    float ss = 0.0f;
    for (int c = 0; c < C_DIM; ++c) { float t = sCov[r * CP + c]; ss += t * t; }
    float second = wave_reduce_add(ss) * (1.0f / C_DIM);
    float num = (C_DIM * trace) * (C_DIM * trace) - second;
    float den = (float)(M_DIM - 1) * (second - trace * trace);
    float rho = fminf(num / den, 1.0f);
    for (int c = 0; c < C_DIM; ++c) {
      float val = (1.0f - rho) * sCov[r * CP + c];
      if (c == r) val += rho * trace + EPSV;
      sCov[r * CP + c] = val;
    }
    const float vinit = v0[(size_t)b * C_DIM + r];
    for (int j = 0; j < C_DIM; ++j) {
      float v = vinit;                      // power iteration restarts from v0
      for (int it = 0; it < NUM_ITER; ++it) {
        sV[r] = v;                          // wave-internal DS ordering suffices
        float w = 0.0f;
        for (int c = 0; c < C_DIM; ++c) w += sCov[r * CP + c] * sV[c];
        float n2 = wave_reduce_add(w * w);
        v = w * __frsqrt_rn(n2);
      }
      sV[r] = v;
      float w = 0.0f;                       // w = cov @ v (unnormalized)
      for (int c = 0; c < C_DIM; ++c) w += sCov[r * CP + c] * sV[c];
      float nume = wave_reduce_add(v * w);
      float deno = wave_reduce_add(v * v);
      float lam = nume / deno;
      sS[j * CP + r] = __frsqrt_rn(lam) * v;        // S_hat row j
      for (int c = 0; c < C_DIM; ++c)               // deflate: cov -= (cov v) v^T
        sCov[r * CP + c] -= w * sV[c];
    }
  }
  __syncthreads();

  // ---- Step E: Z = S_hat @ xc via WMMA, each wave owns 8 N-tiles of 16 ----
  {
    const int rlo = lane & 15;
    const int khi = (lane >> 4) * 2;
    float* __restrict__ zb = z + (size_t)b * (C_DIM * M_DIM);
    for (int nt = 0; nt < 8; ++nt) {
      const int col = (wave * 8 + nt) * 16 + rlo;
      v8f acc0 = {}, acc1 = {};
#pragma unroll
      for (int ks = 0; ks < C_DIM / 4; ++ks) {
        const int kk = ks * 4 + khi;
        v2f a0, a1, bf;
        a0.x = sS[rlo * CP + kk];
        a0.y = sS[rlo * CP + kk + 1];
        a1.x = sS[(rlo + 16) * CP + kk];
        a1.y = sS[(rlo + 16) * CP + kk + 1];
        bf.x = xb[kk * M_DIM + col]       - sMu[kk];
        bf.y = xb[(kk + 1) * M_DIM + col] - sMu[kk + 1];
        acc0 = __builtin_amdgcn_wmma_f32_16x16x4_f32(false, a0, false, bf, (short)0, acc0, false, false);
        acc1 = __builtin_amdgcn_wmma_f32_16x16x4_f32(false, a1, false, bf, (short)0, acc1, false, false);
      }
      const int mrow = (lane >> 4) ? 8 : 0;
#pragma unroll
      for (int i = 0; i < 8; ++i) {
        zb[(mrow + i) * M_DIM + col]      = acc0[i];
        zb[(mrow + i + 16) * M_DIM + col] = acc1[i];
      }
    }
  }
}

extern "C" void kernel_launch(void* const* d_in, const int* in_sizes, int n_in,
                              void* d_out, int out_size, void* d_ws, size_t ws_size,
                              hipStream_t stream) {
  const float* x  = (const float*)d_in[0];
  const float* v0 = (const float*)d_in[1];
  float* z = (float*)d_out;
  const int B = in_sizes[0] / (C_DIM * M_DIM);   // 1024
  pcanorm_kernel<<<dim3(B), dim3(256), 0, stream>>>(x, v0, z);
}